// ScaffoldGINEEncoder_90245852824348
// MI455X (gfx1250) — compile-verified
//
#include <hip/hip_runtime.h>
#include <hip/hip_bf16.h>

// Problem constants (from reference)
#define NN 50000
#define EE 800000
#define GG 1024
#define LL 3
#define DD 96
#define DEE 16
#define DOO 256
#define GN_EPS 1e-5f

static_assert(NN % 16 == 0, "N must tile by 16");
static_assert(EE % 16 == 0, "E must tile by 16");

typedef __attribute__((ext_vector_type(2))) float v2f;
typedef __attribute__((ext_vector_type(8))) float v8f;

// fp32 WMMA: D(16x16) = A(16x4) * B(4x16) + C.  8-arg form:
// (neg_a, A, neg_b, B, c_mod, C, reuse_a, reuse_b)
__device__ __forceinline__ v8f wmma4(v2f a, v2f b, v8f c) {
  return __builtin_amdgcn_wmma_f32_16x16x4_f32(false, a, false, b, (short)0, c,
                                               false, false);
}

__device__ __forceinline__ v8f vzero8() {
  v8f z = {0.f, 0.f, 0.f, 0.f, 0.f, 0.f, 0.f, 0.f};
  return z;
}

// ---------------------------------------------------------------- utilities
__global__ void k_zero(float* __restrict__ p, int n) {
  int i = blockIdx.x * blockDim.x + threadIdx.x;
  if (i < n) p[i] = 0.f;
}

__global__ void k_count(const int* __restrict__ batch, float* __restrict__ cnt) {
  int i = blockIdx.x * blockDim.x + threadIdx.x;
  if (i < NN) atomicAdd(&cnt[batch[i]], 1.f);
}

// ------------------------------------------------------------------- edges
// One wave processes 16 edges: edge_emb = edge_attr[16x16] @ We[16x96] (+be),
// then msg = relu(h[src] + emb) scatter-added into aggr[dst].
__global__ __launch_bounds__(128) void k_edge(
    const float* __restrict__ hin, const float* __restrict__ ea,
    const int* __restrict__ src, const int* __restrict__ dst,
    const float* __restrict__ We, const float* __restrict__ be,
    float* __restrict__ aggr) {
  __shared__ float WeS[DEE * DD];
  __shared__ float beS[DD];
  __shared__ int srcS[4][16];
  __shared__ int dstS[4][16];

  const int tid = threadIdx.x;
  const int w = tid >> 5, lane = tid & 31;
  const int lo = lane & 15, hi = lane >> 4;
  const int e0 = (blockIdx.x * 4 + w) * 16;
  const float* arow = ea + (size_t)(e0 + lo) * DEE;

  // warm the edge_attr tile (global_prefetch_b8) while we fill LDS
  __builtin_prefetch(arow, 0, 0);

  for (int i = tid; i < DEE * DD; i += 128) WeS[i] = We[i];
  if (tid < DD) beS[tid] = be[tid];
  __syncthreads();

  if (lane < 16) {
    srcS[w][lane] = src[e0 + lane];
    dstS[w][lane] = dst[e0 + lane];
  }
  asm volatile("s_wait_dscnt 0" ::: "memory");  // wave-local LDS visibility

  // A tile: 16 edges x 16 features, 4 K-chunks of 4
  v2f a[4];
#pragma unroll
  for (int kc = 0; kc < 4; ++kc) {
    int c = kc * 4 + 2 * hi;
    a[kc] = *(const v2f*)(arow + c);
  }

  int s_m[8], d_m[8];
#pragma unroll
  for (int r = 0; r < 8; ++r) {
    int m = r + 8 * hi;  // D-layout row for accumulator component r
    s_m[r] = srcS[w][m];
    d_m[r] = dstS[w][m];
  }
  // warm gathered h rows while the WMMA chains run
#pragma unroll
  for (int r = 0; r < 8; ++r)
    __builtin_prefetch(hin + (size_t)s_m[r] * DD + lo, 0, 0);

#pragma unroll
  for (int t = 0; t < 6; ++t) {
    const int n = t * 16 + lo;
    // issue all 8 gather loads up front so they overlap the WMMA chain
    float hv[8];
#pragma unroll
    for (int r = 0; r < 8; ++r) hv[r] = hin[(size_t)s_m[r] * DD + n];

    v8f acc = vzero8();
#pragma unroll
    for (int kc = 0; kc < 4; ++kc) {
      int k = kc * 4 + 2 * hi;
      v2f b;
      b.x = WeS[k * DD + n];
      b.y = WeS[(k + 1) * DD + n];
      acc = wmma4(a[kc], b, acc);
    }
    const float bias = beS[n];
#pragma unroll
    for (int r = 0; r < 8; ++r) {
      float v = fmaxf(acc[r] + bias + hv[r], 0.f);
      atomicAdd(&aggr[(size_t)d_m[r] * DD + n], v);
    }
  }
}

// ---------------------------------------------------------------- node MLP
// z = relu((h+aggr)@Wm1+b1)@Wm2+b2 ; write zbuf, accumulate per-graph
// sum(z), sum(z^2) for GraphNorm.  One wave per 16 rows, 4 waves/block.
__global__ __launch_bounds__(128) void k_mlp(
    const float* __restrict__ hin, const float* __restrict__ aggr,
    const float* __restrict__ Wm1, const float* __restrict__ bm1,
    const float* __restrict__ Wm2, const float* __restrict__ bm2,
    const int* __restrict__ batch, float* __restrict__ zbuf,
    float* __restrict__ sumz, float* __restrict__ sumsq) {
  __shared__ float WS[DD * DD];       // 36KB: Wm1 then Wm2
  __shared__ float bS[2][DD];
  __shared__ float z1S[4][16 * DD];   // 24KB staging for intermediate

  const int tid = threadIdx.x;
  const int w = tid >> 5, lane = tid & 31;
  const int lo = lane & 15, hi = lane >> 4;
  const int rt = blockIdx.x * 4 + w;  // 16-row tile index
  const bool active = rt < (NN / 16);
  const int r0 = rt * 16;
  const float* hrow = hin + (size_t)(r0 + lo) * DD;
  const float* grow = aggr + (size_t)(r0 + lo) * DD;

  // warm this wave's A-operand rows while the block stages Wm1 into LDS
  if (active) {
    __builtin_prefetch(hrow, 0, 0);
    __builtin_prefetch(hrow + 32, 0, 0);
    __builtin_prefetch(hrow + 64, 0, 0);
    __builtin_prefetch(grow, 0, 0);
    __builtin_prefetch(grow + 32, 0, 0);
    __builtin_prefetch(grow + 64, 0, 0);
  }

  for (int i = tid; i < DD * DD; i += 128) WS[i] = Wm1[i];
  if (tid < DD) {
    bS[0][tid] = bm1[tid];
    bS[1][tid] = bm2[tid];
  }
  __syncthreads();

  v8f acc[6];
  if (active) {
#pragma unroll
    for (int t = 0; t < 6; ++t) acc[t] = vzero8();
    for (int kc = 0; kc < 24; ++kc) {
      const int c = kc * 4 + 2 * hi;
      v2f hv = *(const v2f*)(hrow + c);
      v2f gv = *(const v2f*)(grow + c);
      v2f a;
      a.x = hv.x + gv.x;
      a.y = hv.y + gv.y;
#pragma unroll
      for (int t = 0; t < 6; ++t) {
        v2f b;
        b.x = WS[c * DD + t * 16 + lo];
        b.y = WS[(c + 1) * DD + t * 16 + lo];
        acc[t] = wmma4(a, b, acc[t]);
      }
    }
    // epilogue 1: bias + relu, stage row-major into LDS
#pragma unroll
    for (int t = 0; t < 6; ++t) {
      const int n = t * 16 + lo;
      const float bias = bS[0][n];
#pragma unroll
      for (int r = 0; r < 8; ++r) {
        float v = fmaxf(acc[t][r] + bias, 0.f);
        z1S[w][(r + 8 * hi) * DD + n] = v;
      }
    }
  }
  __syncthreads();
  for (int i = tid; i < DD * DD; i += 128) WS[i] = Wm2[i];  // swap weights
  __syncthreads();
  if (!active) return;

#pragma unroll
  for (int t = 0; t < 6; ++t) acc[t] = vzero8();
  for (int kc = 0; kc < 24; ++kc) {
    const int c = kc * 4 + 2 * hi;
    v2f a;
    a.x = z1S[w][lo * DD + c];
    a.y = z1S[w][lo * DD + c + 1];
#pragma unroll
    for (int t = 0; t < 6; ++t) {
      v2f b;
      b.x = WS[c * DD + t * 16 + lo];
      b.y = WS[(c + 1) * DD + t * 16 + lo];
      acc[t] = wmma4(a, b, acc[t]);
    }
  }

  int gidx[8];
#pragma unroll
  for (int r = 0; r < 8; ++r) gidx[r] = batch[r0 + r + 8 * hi];

#pragma unroll
  for (int t = 0; t < 6; ++t) {
    const int n = t * 16 + lo;
    const float bias = bS[1][n];
#pragma unroll
    for (int r = 0; r < 8; ++r) {
      const float z = acc[t][r] + bias;
      const int gr = r0 + r + 8 * hi;
      zbuf[(size_t)gr * DD + n] = z;
      atomicAdd(&sumz[gidx[r] * DD + n], z);
      atomicAdd(&sumsq[gidx[r] * DD + n], z * z);
    }
  }
}

// -------------------------------------------------------------- GraphNorm
// var = E[z^2] - 2*ms*mean^2 + ms^2*mean^2 ; h = relu(w*sub*rsqrt(var+eps)+b)
__global__ __launch_bounds__(96) void k_norm(
    const float* __restrict__ zbuf, const int* __restrict__ batch,
    const float* __restrict__ sumz, const float* __restrict__ sumsq,
    const float* __restrict__ cnt, const float* __restrict__ gn_w,
    const float* __restrict__ gn_b, const float* __restrict__ gn_ms,
    float* __restrict__ hout, float* __restrict__ hsum, int last) {
  const int node = blockIdx.x;
  const int d = threadIdx.x;
  const int g = batch[node];
  const float inv = 1.f / fmaxf(cnt[g], 1.f);
  const float mean = sumz[g * DD + d] * inv;
  const float msq = sumsq[g * DD + d] * inv;
  const float ms = gn_ms[d];
  const float var = msq - 2.f * ms * mean * mean + ms * ms * mean * mean;
  const float z = zbuf[(size_t)node * DD + d];
  const float sub = z - ms * mean;
  float v = gn_w[d] * sub * rsqrtf(var + GN_EPS) + gn_b[d];
  v = fmaxf(v, 0.f);
  hout[(size_t)node * DD + d] = v;
  if (last) atomicAdd(&hsum[g * DD + d], v);
}

// ------------------------------------------------------------ final linear
// out[1024x256] = (hsum/cnt) @ lin_W + lin_b, WMMA tiled 16x16.
__global__ __launch_bounds__(128) void k_out(
    const float* __restrict__ hsum, const float* __restrict__ cnt,
    const float* __restrict__ lin_W, const float* __restrict__ lin_b,
    float* __restrict__ out) {
  const int tid = threadIdx.x;
  const int w = tid >> 5, lane = tid & 31;
  const int lo = lane & 15, hi = lane >> 4;
  const int wt = blockIdx.x * 4 + w;  // 64 row tiles * 16 col tiles = 1024
  const int rt = wt >> 4, ct = wt & 15;
  const int row = rt * 16 + lo;
  const float inv = 1.f / fmaxf(cnt[row], 1.f);
  const float* hrow = hsum + (size_t)row * DD;

  v8f acc = vzero8();
  for (int kc = 0; kc < 24; ++kc) {
    const int c = kc * 4 + 2 * hi;
    v2f hv = *(const v2f*)(hrow + c);
    v2f a;
    a.x = hv.x * inv;
    a.y = hv.y * inv;
    v2f b;
    b.x = lin_W[c * DOO + ct * 16 + lo];
    b.y = lin_W[(c + 1) * DOO + ct * 16 + lo];
    acc = wmma4(a, b, acc);
  }
  const float bias = lin_b[ct * 16 + lo];
#pragma unroll
  for (int r = 0; r < 8; ++r)
    out[(size_t)(rt * 16 + r + 8 * hi) * DOO + ct * 16 + lo] = acc[r] + bias;
}

// ------------------------------------------------------------------ driver
extern "C" void kernel_launch(void* const* d_in, const int* in_sizes, int n_in,
                              void* d_out, int out_size, void* d_ws,
                              size_t ws_size, hipStream_t stream) {
  const float* x   = (const float*)d_in[0];
  const float* ea  = (const float*)d_in[1];
  const int* ei    = (const int*)d_in[2];   // [2,E]: row0=src, row1=dst
  const int* batch = (const int*)d_in[3];
  const float* Wm1 = (const float*)d_in[4];
  const float* bm1 = (const float*)d_in[5];
  const float* Wm2 = (const float*)d_in[6];
  const float* bm2 = (const float*)d_in[7];
  const float* We  = (const float*)d_in[8];
  const float* be  = (const float*)d_in[9];
  const float* gnw = (const float*)d_in[10];
  const float* gnb = (const float*)d_in[11];
  const float* gms = (const float*)d_in[12];
  const float* lW  = (const float*)d_in[13];
  const float* lb  = (const float*)d_in[14];
  float* out = (float*)d_out;

  // Workspace layout (floats); needs (3*N*D + 3*G*D + G)*4 ~= 59 MB
  float* ws    = (float*)d_ws;
  float* h     = ws;                      // N*D
  float* zbuf  = h + (size_t)NN * DD;     // N*D
  float* aggr  = zbuf + (size_t)NN * DD;  // N*D
  float* sumz  = aggr + (size_t)NN * DD;  // G*D
  float* sumsq = sumz + GG * DD;          // G*D
  float* hsum  = sumsq + GG * DD;         // G*D
  float* cnt   = hsum + GG * DD;          // G

  const int* src = ei;
  const int* dst = ei + EE;

  // zero hsum + cnt (contiguous), then per-graph node counts
  {
    int n = GG * DD + GG;
    k_zero<<<(n + 255) / 256, 256, 0, stream>>>(hsum, n);
  }
  k_count<<<(NN + 255) / 256, 256, 0, stream>>>(batch, cnt);

  for (int l = 0; l < LL; ++l) {
    // zero aggr + sumz + sumsq (contiguous region)
    {
      int n = NN * DD + 2 * GG * DD;
      k_zero<<<(n + 255) / 256, 256, 0, stream>>>(aggr, n);
    }
    const float* hin = (l == 0) ? x : h;
    k_edge<<<EE / 64, 128, 0, stream>>>(hin, ea, src, dst, We + l * DEE * DD,
                                        be + l * DD, aggr);
    k_mlp<<<(NN / 16 + 3) / 4, 128, 0, stream>>>(
        hin, aggr, Wm1 + l * DD * DD, bm1 + l * DD, Wm2 + l * DD * DD,
        bm2 + l * DD, batch, zbuf, sumz, sumsq);
    k_norm<<<NN, DD, 0, stream>>>(zbuf, batch, sumz, sumsq, cnt, gnw + l * DD,
                                  gnb + l * DD, gms + l * DD, h, hsum,
                                  (l == LL - 1) ? 1 : 0);
  }
  k_out<<<(GG * DOO / 256) / 4, 128, 0, stream>>>(hsum, cnt, lW, lb, out);
}